// SubNonLocalAttention_171798691915
// MI455X (gfx1250) — compile-verified
//
#include <hip/hip_runtime.h>
#include <stdint.h>

// Reference returns (x, y) untouched: the score/top-k math is dead code.
// d_out = [x flat | y flat], pure 512MB-traffic copy -> ~22us at 23.3 TB/s.
// Zero live FLOPs => no WMMA; use the CDNA5 async global<->LDS copy path.

typedef float v4f __attribute__((ext_vector_type(4)));
typedef int   v4i __attribute__((vector_size(4 * sizeof(int))));

#define BLK 256
#define UNROLL 4

#if defined(__HIP_DEVICE_COMPILE__) &&                                   \
    __has_builtin(__builtin_amdgcn_global_load_async_to_lds_b128) &&     \
    __has_builtin(__builtin_amdgcn_global_store_async_from_lds_b128)
#define USE_ASYNC_LDS 1
#else
#define USE_ASYNC_LDS 0
#endif

#if USE_ASYNC_LDS
// Builtin prototype (from the round-1 diagnostic): v4i with AMDGPU address
// spaces — AS(1) global for the memory side, AS(3) for the LDS side.
typedef __attribute__((address_space(1))) v4i gv4i;
typedef __attribute__((address_space(3))) v4i lv4i;

__device__ __forceinline__ void wait_asynccnt0() {
#if __has_builtin(__builtin_amdgcn_s_wait_asynccnt)
  __builtin_amdgcn_s_wait_asynccnt(0);
#else
  asm volatile("s_wait_asynccnt 0" ::: "memory");
#endif
}
#endif

// Copies n floats from src to dst. n4 = n >> 2 handled as 128-bit transfers,
// scalar tail (n & 3) handled by block 0.
__global__ __launch_bounds__(BLK) void hbm_copy_kernel(
    const float* __restrict__ src, float* __restrict__ dst, long long n) {
  const long long n4 = n >> 2;

#if USE_ASYNC_LDS
  // CDNA5 async path: GLOBAL_LOAD_ASYNC_TO_LDS_B128 -> LDS ->
  // GLOBAL_STORE_ASYNC_FROM_LDS_B128, tracked by ASYNCcnt. 16B per lane
  // per issue; UNROLL outstanding transfers per lane per phase.
  __shared__ v4f buf[BLK * UNROLL];  // 16 KB of the 320 KB WGP pool
  const int tid = threadIdx.x;
  const long long chunk = (long long)BLK * UNROLL;            // float4 per block-iter
  const long long gstride = chunk * (long long)gridDim.x;

  for (long long base = (long long)blockIdx.x * chunk; base < n4; base += gstride) {
#pragma unroll
    for (int u = 0; u < UNROLL; ++u) {
      const long long i = base + (long long)u * BLK + tid;
      if (i < n4) {
        // generic->AS1 is value-preserving; generic->AS3 is the low-32-bit
        // truncation per the LDS aperture rule (ISA 10.2).
        gv4i* g = (gv4i*)(uintptr_t)(src + i * 4);
        lv4i* l = (lv4i*)(uint32_t)(uintptr_t)(&buf[u * BLK + tid]);
        __builtin_amdgcn_global_load_async_to_lds_b128(g, l, 0, 0);
      }
    }
    wait_asynccnt0();  // loads done before stores read LDS
#pragma unroll
    for (int u = 0; u < UNROLL; ++u) {
      const long long i = base + (long long)u * BLK + tid;
      if (i < n4) {
        gv4i* g = (gv4i*)(uintptr_t)(dst + i * 4);
        lv4i* l = (lv4i*)(uint32_t)(uintptr_t)(&buf[u * BLK + tid]);
        __builtin_amdgcn_global_store_async_from_lds_b128(g, l, 0, 0);
      }
    }
    wait_asynccnt0();  // stores drained before LDS buffer reuse
  }
#else
  // Fallback: direct 128-bit non-temporal streaming copy (working set
  // >> 192MB L2, so bypass-style temporal hints are correct).
  const v4f* __restrict__ s4 = (const v4f*)src;
  v4f* __restrict__ d4 = (v4f*)dst;
  const long long stride = (long long)gridDim.x * BLK;
  for (long long i = (long long)blockIdx.x * BLK + threadIdx.x; i < n4; i += stride) {
    v4f v = __builtin_nontemporal_load(&s4[i]);
    __builtin_nontemporal_store(v, &d4[i]);
  }
#endif

  // Scalar tail (not hit for these shapes: counts are multiples of 4).
  if (blockIdx.x == 0 && (long long)threadIdx.x < (n & 3LL)) {
    const long long t = (n & ~3LL) + threadIdx.x;
    dst[t] = src[t];
  }
}

extern "C" void kernel_launch(void* const* d_in, const int* in_sizes, int n_in,
                              void* d_out, int out_size, void* d_ws, size_t ws_size,
                              hipStream_t stream) {
  (void)n_in; (void)out_size; (void)d_ws; (void)ws_size;

  const float* x = (const float*)d_in[0];   // (128,64,64,64) f32
  const float* y = (const float*)d_in[1];   // (128,64,64,64) f32
  float* out = (float*)d_out;               // [x | y] flat

  const long long nx = in_sizes[0];
  const long long ny = in_sizes[1];

  const long long chunk = (long long)BLK * UNROLL;      // float4 per block-iter
  const long long nx4 = nx >> 2, ny4 = ny >> 2;
  int gx = (int)((nx4 + chunk - 1) / chunk); if (gx < 1) gx = 1;
  int gy = (int)((ny4 + chunk - 1) / chunk); if (gy < 1) gy = 1;

  hbm_copy_kernel<<<gx, BLK, 0, stream>>>(x, out, nx);
  hbm_copy_kernel<<<gy, BLK, 0, stream>>>(y, out + nx, ny);
}